// ODEBlock_15178414424744
// MI455X (gfx1250) — compile-verified
//
#include <hip/hip_runtime.h>
#include <math.h>

// ---------------- types ----------------
typedef __attribute__((ext_vector_type(2))) float v2f;
typedef __attribute__((ext_vector_type(8))) float v8f;

struct OdeState { float t; float h; int accept; int pad; };

// ---------------- problem constants ----------------
#define NROWS   65536
#define NDIM    64
#define TILE_M  64
#define NTHREADS 256
#define EPT     16                 // (TILE_M*NDIM)/NTHREADS
#define LSTRIDE 65                 // padded LDS row stride (floats)
#define KSZ     (TILE_M * LSTRIDE) // one 64x64 padded tile in LDS
#define ENDTIME 5.0f
#define RTOLC   1e-5f
#define ATOLC   1e-5f
#define H0C     0.01f

// Dormand-Prince 5(4) tableau (full 6x6 A, zero padded)
__device__ constexpr float cA[6][6] = {
  {1.f/5.f, 0.f, 0.f, 0.f, 0.f, 0.f},
  {3.f/40.f, 9.f/40.f, 0.f, 0.f, 0.f, 0.f},
  {44.f/45.f, -56.f/15.f, 32.f/9.f, 0.f, 0.f, 0.f},
  {19372.f/6561.f, -25360.f/2187.f, 64448.f/6561.f, -212.f/729.f, 0.f, 0.f},
  {9017.f/3168.f, -355.f/33.f, 46732.f/5247.f, 49.f/176.f, -5103.f/18656.f, 0.f},
  {35.f/384.f, 0.f, 500.f/1113.f, 125.f/192.f, -2187.f/6784.f, 11.f/84.f},
};
__device__ constexpr float cB5[7] = {35.f/384.f, 0.f, 500.f/1113.f, 125.f/192.f,
                                     -2187.f/6784.f, 11.f/84.f, 0.f};
__device__ constexpr float cE[7]  = {71.f/57600.f, 0.f, -71.f/16695.f, 71.f/1920.f,
                                     -17253.f/339200.f, 22.f/525.f, -1.f/40.f};

// ---- one 16x16 output tile of  K = sin(-(Ys @ W^T + b))  via fp32 WMMA ----
// sYs : [64][LSTRIDE]  (row-major y_stage)     A = Ys (16x4 slices)
// sWt : [64][LSTRIDE]  sWt[k][j] = W[j][k]     B = W^T (4x16 slices)
__device__ __forceinline__ void gemm_sin_tile(const float* __restrict__ sYs,
                                              const float* __restrict__ sWt,
                                              const float* __restrict__ sB,
                                              float* __restrict__ sKdst,
                                              int tileIdx, int lane) {
  const int mt   = tileIdx >> 2;
  const int nt   = tileIdx & 3;
  const int half = lane >> 4;     // 0: lanes 0-15, 1: lanes 16-31
  const int l15  = lane & 15;

  v8f acc = {};
  const int am = mt * 16 + l15;   // A-frag row (M)
  const int bn = nt * 16 + l15;   // B-frag col (N)
#pragma unroll
  for (int k0 = 0; k0 < NDIM; k0 += 4) {
    const int ka = k0 + half * 2; // K pair for this half-wave
    v2f a, bfr;
    a.x   = sYs[am * LSTRIDE + ka];
    a.y   = sYs[am * LSTRIDE + ka + 1];
    bfr.x = sWt[ka * LSTRIDE + bn];
    bfr.y = sWt[(ka + 1) * LSTRIDE + bn];
    acc = __builtin_amdgcn_wmma_f32_16x16x4_f32(false, a, false, bfr,
                                                (short)0, acc, false, false);
  }
  const float bias = sB[bn];
#pragma unroll
  for (int r = 0; r < 8; ++r) {
    const int m = mt * 16 + half * 8 + r;   // D-frag row layout
    sKdst[m * LSTRIDE + bn] = sinf(-(acc[r] + bias));
  }
}

// ---------------- fused DOPRI5 step (one RK step for all rows) ----------------
extern "C" __global__ void __launch_bounds__(NTHREADS)
dopri5_step(float* __restrict__ y, float* __restrict__ ynew,
            const float* __restrict__ W, const float* __restrict__ bvec,
            const OdeState* __restrict__ st, float* __restrict__ errsum) {
  extern __shared__ float smem[];
  float* sWt  = smem;                  // 64*65
  float* sB   = sWt + NDIM * LSTRIDE;  // 64
  float* sYs  = sB + NDIM;             // 64*65
  float* sK   = sYs + KSZ;             // 7 * 64*65
  float* sRed = sK + 7 * KSZ;          // 256

  const int tid  = threadIdx.x;
  const int lane = tid & 31;
  const int wave = tid >> 5;

  // uniform scalar state from previous decide
  const float t   = st->t;
  const float h   = st->h;
  const int   acc = st->accept;
  const float rem   = ENDTIME - t;
  const float h_eff = fminf(h, fmaxf(rem, 1e-12f));

  // stage W^T and b into LDS
  for (int i = tid; i < NDIM * NDIM; i += NTHREADS) {
    const int j = i >> 6, k = i & 63;
    sWt[k * LSTRIDE + j] = W[j * NDIM + k];
  }
  if (tid < NDIM) sB[tid] = bvec[tid];

  // load base tile (commit previous accept), keep in registers
  const int base = blockIdx.x * (TILE_M * NDIM);
  const float* __restrict__ src = acc ? ynew : y;
  float yb[EPT];
#pragma unroll
  for (int r = 0; r < EPT; ++r) yb[r] = src[base + r * NTHREADS + tid];
  if (acc) {
#pragma unroll
    for (int r = 0; r < EPT; ++r) y[base + r * NTHREADS + tid] = yb[r];
  }

  // ---- stage 0: k0 = f(y) ----
#pragma unroll
  for (int r = 0; r < EPT; ++r) {
    const int e = r * NTHREADS + tid;
    sYs[(e >> 6) * LSTRIDE + (e & 63)] = yb[r];
  }
  __syncthreads();
  gemm_sin_tile(sYs, sWt, sB, sK, wave, lane);
  gemm_sin_tile(sYs, sWt, sB, sK, wave + 8, lane);
  __syncthreads();

  // ---- stages 1..6 ----
#pragma unroll
  for (int i = 1; i < 7; ++i) {
#pragma unroll
    for (int r = 0; r < EPT; ++r) {
      const int e  = r * NTHREADS + tid;
      const int la = (e >> 6) * LSTRIDE + (e & 63);
      float s = 0.f;
#pragma unroll
      for (int j = 0; j < 6; ++j)
        if (j < i && cA[i - 1][j] != 0.f) s += cA[i - 1][j] * sK[j * KSZ + la];
      sYs[la] = yb[r] + h_eff * s;
    }
    __syncthreads();
    gemm_sin_tile(sYs, sWt, sB, sK + i * KSZ, wave, lane);
    gemm_sin_tile(sYs, sWt, sB, sK + i * KSZ, wave + 8, lane);
    __syncthreads();
  }

  // ---- combine: y5, embedded error, squared-norm partial ----
  float local = 0.f;
#pragma unroll
  for (int r = 0; r < EPT; ++r) {
    const int e  = r * NTHREADS + tid;
    const int la = (e >> 6) * LSTRIDE + (e & 63);
    float sy = 0.f, se = 0.f;
#pragma unroll
    for (int j = 0; j < 7; ++j) {
      const float kj = sK[j * KSZ + la];
      if (cB5[j] != 0.f) sy += cB5[j] * kj;
      if (cE[j]  != 0.f) se += cE[j]  * kj;
    }
    const float y5 = yb[r] + h_eff * sy;
    const float er = h_eff * se;
    ynew[base + e] = y5;
    const float scale = ATOLC + RTOLC * fmaxf(fabsf(yb[r]), fabsf(y5));
    const float q = er / scale;
    local += q * q;
  }

  // block reduction -> atomic accumulate
  sRed[tid] = local;
  __syncthreads();
  for (int s = NTHREADS / 2; s > 0; s >>= 1) {
    if (tid < s) sRed[tid] += sRed[tid + s];
    __syncthreads();
  }
  if (tid == 0) atomicAdd(errsum, sRed[0]);
}

// ---------------- scalar controller ----------------
extern "C" __global__ void dopri5_decide(OdeState* st, float* errsum) {
  if (threadIdx.x == 0 && blockIdx.x == 0) {
    const float t = st->t, h = st->h;
    const float en  = sqrtf(*errsum / (float)(NROWS * NDIM));
    const float rem = ENDTIME - t;
    const bool  done   = rem <= 0.f;
    const float h_eff  = fminf(h, fmaxf(rem, 1e-12f));
    const bool  accept = (en <= 1.f) && !done;
    st->accept = accept ? 1 : 0;
    if (accept) st->t = t + h_eff;
    float fac = 0.9f * powf(fmaxf(en, 1e-10f), -0.2f);
    fac = fminf(fmaxf(fac, 0.2f), 10.f);
    st->h = done ? h : h_eff * fac;
    *errsum = 0.f;
  }
}

extern "C" __global__ void dopri5_init(OdeState* st, float* errsum) {
  if (threadIdx.x == 0 && blockIdx.x == 0) {
    st->t = 0.f; st->h = H0C; st->accept = 0; st->pad = 0;
    *errsum = 0.f;
  }
}

// apply final accept decision: y = accept ? ynew : y   (float4 vectorized)
extern "C" __global__ void __launch_bounds__(NTHREADS)
dopri5_final(float* __restrict__ y, const float* __restrict__ ynew,
             const OdeState* __restrict__ st) {
  if (!st->accept) return;
  const int i = blockIdx.x * NTHREADS + threadIdx.x;
  ((float4*)y)[i] = ((const float4*)ynew)[i];
}

// ---------------- host launcher ----------------
extern "C" void kernel_launch(void* const* d_in, const int* in_sizes, int n_in,
                              void* d_out, int out_size, void* d_ws, size_t ws_size,
                              hipStream_t stream) {
  (void)in_sizes; (void)n_in; (void)out_size; (void)ws_size;
  const float* x = (const float*)d_in[0];
  const float* W = (const float*)d_in[1];
  const float* b = (const float*)d_in[2];

  float* y    = (float*)d_out;                       // committed state (16 MB)
  float* ynew = (float*)d_ws;                        // candidate state (16 MB)
  char*  wsb  = (char*)d_ws;
  OdeState* st  = (OdeState*)(wsb + (size_t)NROWS * NDIM * sizeof(float));
  float* errsum = (float*)(st + 1);

  const size_t ybytes = (size_t)NROWS * NDIM * sizeof(float);
  hipMemcpyAsync(y, x, ybytes, hipMemcpyDeviceToDevice, stream);
  hipLaunchKernelGGL(dopri5_init, dim3(1), dim3(1), 0, stream, st, errsum);

  const size_t smem = (size_t)(NDIM * LSTRIDE   // Wt
                               + NDIM           // b
                               + KSZ            // Ys
                               + 7 * KSZ        // k0..k6
                               + NTHREADS)      // reduction
                      * sizeof(float);          // = 151,040 B (< 160 KB)
  const dim3 grid(NROWS / TILE_M);              // 1024 blocks
  const dim3 blk(NTHREADS);                     // 8 waves (wave32)

  for (int it = 0; it < 128; ++it) {
    hipLaunchKernelGGL(dopri5_step, grid, blk, smem, stream,
                       y, ynew, W, b, st, errsum);
    hipLaunchKernelGGL(dopri5_decide, dim3(1), dim3(1), 0, stream, st, errsum);
  }
  hipLaunchKernelGGL(dopri5_final, dim3((NROWS * NDIM / 4) / NTHREADS), blk,
                     0, stream, y, ynew, st);
}